// GNNLayer_7756710937144
// MI455X (gfx1250) — compile-verified
//
#include <hip/hip_runtime.h>
#include <hip/hip_bf16.h>
#include <math.h>

#define IN_DIM   64
#define OUT_DIM  64
#define ATTN_DIM 5
#define N_NODE   500000
#define N_EDGE   1000000

typedef __attribute__((ext_vector_type(2))) float v2f;
typedef __attribute__((ext_vector_type(8))) float v8f;

// Float atomic max via monotonic int reinterpretation:
//  val >= 0 -> signed atomicMax ; val < 0 -> unsigned atomicMin.
// Works with -inf (0xFF800000) initialization.
__device__ __forceinline__ void atomic_max_float(float* addr, float val) {
    if (val >= 0.0f)
        atomicMax((int*)addr, __float_as_int(val));
    else
        atomicMin((unsigned int*)addr, __float_as_uint(val));
}

// ---------------------------------------------------------------------------
// K0: init agg = -inf, out = 0, reduction scalars {max=-inf, sum=0}
// ---------------------------------------------------------------------------
__global__ void k_init(float* __restrict__ agg, float* __restrict__ out,
                       float* __restrict__ red, int n) {
    int i = blockIdx.x * blockDim.x + threadIdx.x;
    int stride = gridDim.x * blockDim.x;
    for (; i < n; i += stride) { agg[i] = -INFINITY; out[i] = 0.0f; }
    if (blockIdx.x == 0 && threadIdx.x == 0) { red[0] = -INFINITY; red[1] = 0.0f; }
}

// ---------------------------------------------------------------------------
// K1: per-edge attention + alpha + atomic segment-max of alpha*(hs-hr).
// One wave32 per edge; lane l covers dims 2l, 2l+1 (coalesced float2 rows).
// ---------------------------------------------------------------------------
__global__ void k_edge_attn(const float* __restrict__ hidden,
                            const float* __restrict__ rela,
                            const float* __restrict__ Ws,
                            const float* __restrict__ Wr,
                            const float* __restrict__ Wqr,
                            const float* __restrict__ Wqr_b,
                            const float* __restrict__ walpha_w,
                            const float* __restrict__ walpha_b,
                            const int*   __restrict__ q_rel,
                            const int*   __restrict__ edges,
                            float* __restrict__ agg) {
    const int lane = threadIdx.x & 31;
    const int wid  = (blockIdx.x * blockDim.x + threadIdx.x) >> 5;
    const int nw   = (gridDim.x * blockDim.x) >> 5;
    const int d0   = 2 * lane;

    for (int e = wid; e < N_EDGE; e += nw) {
        const int r_idx = edges[e * 6 + 0];
        const int rel   = edges[e * 6 + 2];
        const int sub   = edges[e * 6 + 4];
        const int obj   = edges[e * 6 + 5];
        const int qr    = q_rel[r_idx];

        v2f hs = *(const v2f*)(hidden + (long long)sub * 64 + d0);
        v2f hr = *(const v2f*)(rela   + (long long)rel * 64 + d0);
        v2f hq = *(const v2f*)(rela   + (long long)qr  * 64 + d0);

        float p[ATTN_DIM];
#pragma unroll
        for (int j = 0; j < ATTN_DIM; ++j) {
            p[j] = hs.x * Ws[d0 * 5 + j]  + hs.y * Ws[(d0 + 1) * 5 + j]
                 + hr.x * Wr[d0 * 5 + j]  + hr.y * Wr[(d0 + 1) * 5 + j]
                 + hq.x * Wqr[d0 * 5 + j] + hq.y * Wqr[(d0 + 1) * 5 + j];
        }
        // full-wave butterfly reduction (wave32)
#pragma unroll
        for (int off = 16; off >= 1; off >>= 1) {
#pragma unroll
            for (int j = 0; j < ATTN_DIM; ++j)
                p[j] += __shfl_xor(p[j], off, 32);
        }
        float s = walpha_b[0];
#pragma unroll
        for (int j = 0; j < ATTN_DIM; ++j) {
            float a = p[j] + Wqr_b[j];
            a = a > 0.0f ? a : 0.0f;            // relu
            s += a * walpha_w[j];
        }
        float alpha = 1.0f / (1.0f + __expf(-s)); // sigmoid

        float mx = alpha * (hs.x - hr.x);
        float my = alpha * (hs.y - hr.y);
        float* dst = agg + (long long)obj * 64 + d0;
        atomic_max_float(dst,     mx);
        atomic_max_float(dst + 1, my);
    }
}

// ---------------------------------------------------------------------------
// K2: hidden_new = where(agg==-inf,0,agg) @ Wh, in place, fp32 WMMA 16x16x4.
// One wave per 16 rows; 4 N-tiles x 16 K-steps of V_WMMA_F32_16X16X4_F32.
// A layout: lane<16 holds A[M=lane][K=4s+0,1]; lane>=16 holds K=4s+2,3.
// B layout: bx = B[K=4s+2*hi][N], by = B[K=4s+2*hi+1][N], N = 16t+lo.
// D layout: vgpr r -> M = r + 8*hi, N = 16t+lo.
// ---------------------------------------------------------------------------
__global__ void k_gemm_wh(float* __restrict__ hn, const float* __restrict__ Wh) {
    const int lane = threadIdx.x & 31;
    const int wid  = (blockIdx.x * blockDim.x + threadIdx.x) >> 5;
    if (wid >= N_NODE / 16) return;          // wave-uniform: EXEC stays all-1s
    const int hi = lane >> 4, lo = lane & 15;
    const long long base = (long long)wid * 16;

    v2f a[16];
#pragma unroll
    for (int s = 0; s < 16; ++s) {
        v2f v = *(const v2f*)(hn + (base + lo) * 64 + 4 * s + 2 * hi);
        v.x = (v.x == -INFINITY) ? 0.0f : v.x;   // jnp.where(isneginf, 0, agg)
        v.y = (v.y == -INFINITY) ? 0.0f : v.y;
        a[s] = v;
    }
#pragma unroll
    for (int t = 0; t < 4; ++t) {
        v8f acc = {};
#pragma unroll
        for (int s = 0; s < 16; ++s) {
            v2f b;
            b.x = Wh[(4 * s + 2 * hi)     * 64 + 16 * t + lo];
            b.y = Wh[(4 * s + 2 * hi + 1) * 64 + 16 * t + lo];
            acc = __builtin_amdgcn_wmma_f32_16x16x4_f32(
                false, a[s], false, b, (short)0, acc, false, false);
        }
#pragma unroll
        for (int r = 0; r < 8; ++r)
            hn[(base + r + 8 * hi) * 64 + 16 * t + lo] = acc[r];
    }
}

// ---------------------------------------------------------------------------
// K3: per-edge score = leaky_relu([hn[sub],hn[obj]] @ fc_w + b), global max.
// ---------------------------------------------------------------------------
__global__ void k_score(const float* __restrict__ hn,
                        const float* __restrict__ fcw,
                        const float* __restrict__ fcb,
                        const int*   __restrict__ edges,
                        float* __restrict__ scores,
                        float* __restrict__ red) {
    const int lane = threadIdx.x & 31;
    const int wid  = (blockIdx.x * blockDim.x + threadIdx.x) >> 5;
    const int nw   = (gridDim.x * blockDim.x) >> 5;
    const int d0   = 2 * lane;
    float lmax = -INFINITY;

    for (int e = wid; e < N_EDGE; e += nw) {
        const int sub = edges[e * 6 + 4];
        const int obj = edges[e * 6 + 5];
        v2f a = *(const v2f*)(hn + (long long)sub * 64 + d0);
        v2f b = *(const v2f*)(hn + (long long)obj * 64 + d0);
        float p = a.x * fcw[d0] + a.y * fcw[d0 + 1]
                + b.x * fcw[64 + d0] + b.y * fcw[64 + d0 + 1];
#pragma unroll
        for (int off = 16; off >= 1; off >>= 1) p += __shfl_xor(p, off, 32);
        float sc = p + fcb[0];
        sc = sc > 0.0f ? sc : 0.2f * sc;         // leaky_relu(0.2)
        if (lane == 0) scores[e] = sc;
        lmax = fmaxf(lmax, sc);
    }
    if (lane == 0) atomic_max_float(red, lmax);
}

// ---------------------------------------------------------------------------
// K4: scores[e] = exp(scores[e]-gmax); accumulate global sum.
// ---------------------------------------------------------------------------
__global__ void k_expsum(float* __restrict__ scores, float* __restrict__ red) {
    const float gmax = red[0];
    float lsum = 0.0f;
    int i = blockIdx.x * blockDim.x + threadIdx.x;
    int stride = gridDim.x * blockDim.x;
    for (int e = i; e < N_EDGE; e += stride) {
        float v = __expf(scores[e] - gmax);
        scores[e] = v;
        lsum += v;
    }
#pragma unroll
    for (int off = 16; off >= 1; off >>= 1) lsum += __shfl_xor(lsum, off, 32);
    if ((threadIdx.x & 31) == 0) atomicAdd(red + 1, lsum);
}

// ---------------------------------------------------------------------------
// K5: per 16 edges: D = hn[obj] @ Wnode (fp32 WMMA), then
//     atomicAdd(out[sub], w_e * (D + b)).
// ---------------------------------------------------------------------------
__global__ void k_final(const float* __restrict__ hn,
                        const float* __restrict__ Wnode,
                        const float* __restrict__ Wnode_b,
                        const int*   __restrict__ edges,
                        const float* __restrict__ scores,
                        const float* __restrict__ red,
                        float* __restrict__ out) {
    const int lane = threadIdx.x & 31;
    const int wid  = (blockIdx.x * blockDim.x + threadIdx.x) >> 5;
    if (wid >= N_EDGE / 16) return;          // wave-uniform exit
    const int hi = lane >> 4, lo = lane & 15;
    const int base = wid * 16;
    const float invsum = 1.0f / red[1];

    const int rowA = edges[(base + lo) * 6 + 5];   // obj of edge (base+lo)
    v2f a[16];
#pragma unroll
    for (int s = 0; s < 16; ++s)
        a[s] = *(const v2f*)(hn + (long long)rowA * 64 + 4 * s + 2 * hi);

    float wv[8]; int sv[8];
#pragma unroll
    for (int r = 0; r < 8; ++r) {
        const int eidx = base + 8 * hi + r;        // edge for D-row M=r+8*hi
        wv[r] = scores[eidx] * invsum;             // softmax weight
        sv[r] = edges[eidx * 6 + 4];               // sub (scatter target)
    }
#pragma unroll
    for (int t = 0; t < 4; ++t) {
        v8f acc = {};
#pragma unroll
        for (int s = 0; s < 16; ++s) {
            v2f b;
            b.x = Wnode[(4 * s + 2 * hi)     * 64 + 16 * t + lo];
            b.y = Wnode[(4 * s + 2 * hi + 1) * 64 + 16 * t + lo];
            acc = __builtin_amdgcn_wmma_f32_16x16x4_f32(
                false, a[s], false, b, (short)0, acc, false, false);
        }
        const int n  = 16 * t + lo;
        const float bn = Wnode_b[n];
#pragma unroll
        for (int r = 0; r < 8; ++r) {
            float val = wv[r] * (acc[r] + bn);
            atomicAdd(out + (long long)sv[r] * 64 + n, val);
        }
    }
}

// ---------------------------------------------------------------------------
extern "C" void kernel_launch(void* const* d_in, const int* in_sizes, int n_in,
                              void* d_out, int out_size, void* d_ws, size_t ws_size,
                              hipStream_t stream) {
    const float* hidden   = (const float*)d_in[0];
    const float* rela     = (const float*)d_in[1];
    const float* Ws       = (const float*)d_in[2];
    const float* Wr       = (const float*)d_in[3];
    const float* Wqr_w    = (const float*)d_in[4];
    const float* Wqr_b    = (const float*)d_in[5];
    const float* walpha_w = (const float*)d_in[6];
    const float* walpha_b = (const float*)d_in[7];
    const float* Wh       = (const float*)d_in[8];
    const float* fc_w     = (const float*)d_in[9];
    const float* fc_b     = (const float*)d_in[10];
    const float* Wnode_w  = (const float*)d_in[11];
    const float* Wnode_b  = (const float*)d_in[12];
    const int*   q_rel    = (const int*)d_in[14];
    const int*   edges    = (const int*)d_in[15];
    float* out = (float*)d_out;

    // Workspace layout: agg/hidden_new (in-place GEMM) | scores | 2 reduction floats
    float* agg    = (float*)d_ws;                     // N_NODE*64 floats (128 MB)
    float* scores = agg + (size_t)N_NODE * 64;        // N_EDGE floats (4 MB)
    float* red    = scores + N_EDGE;                  // {max, sum}

    k_init<<<2048, 256, 0, stream>>>(agg, out, red, N_NODE * 64);
    k_edge_attn<<<4096, 256, 0, stream>>>(hidden, rela, Ws, Wr, Wqr_w, Wqr_b,
                                          walpha_w, walpha_b, q_rel, edges, agg);
    k_gemm_wh<<<(N_NODE / 16 + 7) / 8, 256, 0, stream>>>(agg, Wh);
    k_score<<<4096, 256, 0, stream>>>(agg, fc_w, fc_b, edges, scores, red);
    k_expsum<<<4096, 256, 0, stream>>>(scores, red);
    k_final<<<(N_EDGE / 16 + 7) / 8, 256, 0, stream>>>(agg, Wnode_w, Wnode_b,
                                                       edges, scores, red, out);
}